// EdgeNetwork_24524263260404
// MI455X (gfx1250) — compile-verified
//
#include <hip/hip_runtime.h>

typedef __attribute__((ext_vector_type(2))) float v2f;
typedef __attribute__((ext_vector_type(8))) float v8f;

#define HID  8
#define EPS  1e-5f

// D = A(16x4 f32) * B(4x16 f32) + C(16x16 f32)
#define WMMA_F32X4(A, B, C) \
    __builtin_amdgcn_wmma_f32_16x16x4_f32(false, (A), false, (B), (short)0, (C), false, false)

__device__ __forceinline__ float tanh_fast(float x) {
#if __has_builtin(__builtin_amdgcn_tanhf)
    return __builtin_amdgcn_tanhf(x);   // gfx1250 v_tanh_f32
#else
    // monotone, saturates correctly at +/-1 without NaN for large |x|
    return 1.0f - 2.0f / (__expf(2.0f * x) + 1.0f);
#endif
}

// A-fragment for W^T placed in D-rows [mbase, mbase+8), chunk starting at K=k0.
// A layout (16x4 f32): lanes 0-15: row M=lane,    v0=K=k0,   v1=K=k0+1
//                      lanes 16-31: row M=lane-16, v0=K=k0+2, v1=K=k0+3
// W row-major [fan_in][8]; W^T[m][k] = W[k*8+m].
__device__ __forceinline__ v2f load_a_frag(const float* __restrict__ W,
                                           int fan_in, int k0, int mbase, int lane) {
    const int m  = (lane & 15) - mbase;        // valid feature row if 0..7
    const int hi = lane >> 4;
    const int ka = k0 + 2 * hi;
    const int kb = ka + 1;
    const bool mok = (m >= 0) && (m < HID);
    v2f a;
    a.x = (mok && ka < fan_in) ? W[ka * HID + m] : 0.0f;
    a.y = (mok && kb < fan_in) ? W[kb * HID + m] : 0.0f;
    return a;
}

// bias + layernorm + tanh, per-lane over the 8 accumulator registers.
// All 32 lanes carry a live edge.
__device__ __forceinline__ void ln_tanh(const v8f& c,
                                        const float* __restrict__ bias,
                                        const float* __restrict__ g,
                                        const float* __restrict__ be,
                                        float h[8]) {
    float t[8];
    float s = 0.0f;
#pragma unroll
    for (int f = 0; f < 8; ++f) { t[f] = c[f] + bias[f]; s += t[f]; }
    const float mean = s * 0.125f;
    float var = 0.0f;
#pragma unroll
    for (int f = 0; f < 8; ++f) { const float d = t[f] - mean; var += d * d; }
    var *= 0.125f;
    const float r = rsqrtf(var + EPS);
#pragma unroll
    for (int f = 0; f < 8; ++f)
        h[f] = tanh_fast((t[f] - mean) * r * g[f] + be[f]);
}

// Hidden-layer B fragments for both tiles, chunk K0.
// h[f] = feature f of edge (base+lane), live in ALL lanes.
// Tile A = edges in lanes 0-15, tile B = edges in lanes 16-31.
template <int K0>
__device__ __forceinline__ void hidden_b_frags(const float h[8], int lane,
                                               v2f& bA, v2f& bB) {
    const int  n  = lane & 15;
    const bool lo = lane < 16;
    // tile A, hi-lane rows K0+2/K0+3 come from lanes 0-15
    const float a2 = __shfl(h[K0 + 2], n, 32);
    const float a3 = __shfl(h[K0 + 3], n, 32);
    // tile B, lo-lane rows K0/K0+1 come from lanes 16-31
    const float b0 = __shfl(h[K0],     n + 16, 32);
    const float b1 = __shfl(h[K0 + 1], n + 16, 32);
    bA.x = lo ? h[K0]     : a2;
    bA.y = lo ? h[K0 + 1] : a3;
    bB.x = lo ? b0 : h[K0 + 2];
    bB.y = lo ? b1 : h[K0 + 3];
}

__global__ void __launch_bounds__(256)
edge_mlp_wmma_kernel(const float* __restrict__ x,
                     const long long* __restrict__ edge_index,   // [2, E] int64
                     const float* __restrict__ W0, const float* __restrict__ b0,
                     const float* __restrict__ g0, const float* __restrict__ be0,
                     const float* __restrict__ W1, const float* __restrict__ b1,
                     const float* __restrict__ g1, const float* __restrict__ be1,
                     const float* __restrict__ W2, const float* __restrict__ b2,
                     const float* __restrict__ g2, const float* __restrict__ be2,
                     const float* __restrict__ W3, const float* __restrict__ b3v,
                     float* __restrict__ out, int n_edges) {
    const int lane          = threadIdx.x & 31;
    const int wave          = threadIdx.x >> 5;
    const int wavesPerBlock = blockDim.x >> 5;
    const int gwave         = blockIdx.x * wavesPerBlock + wave;
    const int totalWaves    = gridDim.x * wavesPerBlock;
    const int nIters        = (n_edges + 31) >> 5;   // 32 edges per wave-iter

    // Hoisted weight fragments: [layer][row-half][K-chunk]
    const v2f a0l0 = load_a_frag(W0, 6, 0, 0, lane);
    const v2f a0l4 = load_a_frag(W0, 6, 4, 0, lane);
    const v2f a0h0 = load_a_frag(W0, 6, 0, 8, lane);
    const v2f a0h4 = load_a_frag(W0, 6, 4, 8, lane);
    const v2f a1l0 = load_a_frag(W1, 8, 0, 0, lane);
    const v2f a1l4 = load_a_frag(W1, 8, 4, 0, lane);
    const v2f a1h0 = load_a_frag(W1, 8, 0, 8, lane);
    const v2f a1h4 = load_a_frag(W1, 8, 4, 8, lane);
    const v2f a2l0 = load_a_frag(W2, 8, 0, 0, lane);
    const v2f a2l4 = load_a_frag(W2, 8, 4, 0, lane);
    const v2f a2h0 = load_a_frag(W2, 8, 0, 8, lane);
    const v2f a2h4 = load_a_frag(W2, 8, 4, 8, lane);

    const int  n  = lane & 15;
    const bool lo = lane < 16;

    for (int it = gwave; it < nIters; it += totalWaves) {
        const int base = it * 32;
        const int e    = base + lane;                 // one edge per lane
        const int ec   = (e < n_edges) ? e : (n_edges - 1);

        // Stream indices non-temporally (read exactly once); keep x hot in L2.
        const long long sN = __builtin_nontemporal_load(edge_index + ec);
        const long long tN = __builtin_nontemporal_load(edge_index + n_edges + ec);
        const float* xs = x + sN * 3;
        const float* xt = x + tN * 3;
        const float xs0 = xs[0], xs1 = xs[1], xs2 = xs[2];
        const float xe0 = xt[0], xe1 = xt[1], xe2 = xt[2];

        // Layer-0 B fragments. Edge feature vector: [xs0 xs1 xs2 xe0 xe1 xe2 0 0]
        // Tile A (edges base..base+15, data in lanes 0-15):
        const float tA_xs2 = __shfl(xs2, n, 32);
        const float tA_xe0 = __shfl(xe0, n, 32);
        // Tile B (edges base+16..base+31, data in lanes 16-31):
        const float tB_xs0 = __shfl(xs0, n + 16, 32);
        const float tB_xs1 = __shfl(xs1, n + 16, 32);
        const float tB_xe1 = __shfl(xe1, n + 16, 32);
        const float tB_xe2 = __shfl(xe2, n + 16, 32);

        v2f bA0, bA4, bB0, bB4;
        bA0.x = lo ? xs0 : tA_xs2;     // K=0 | K=2
        bA0.y = lo ? xs1 : tA_xe0;     // K=1 | K=3
        bA4.x = lo ? xe1 : 0.0f;       // K=4 | K=6 (pad)
        bA4.y = lo ? xe2 : 0.0f;       // K=5 | K=7 (pad)
        bB0.x = lo ? tB_xs0 : xs2;
        bB0.y = lo ? tB_xs1 : xe0;
        bB4.x = lo ? tB_xe1 : 0.0f;
        bB4.y = lo ? tB_xe2 : 0.0f;

        // Layer 0: 6 -> 8, both 16-edge tiles accumulated into one C
        v8f c = {};
        c = WMMA_F32X4(a0l0, bA0, c);
        c = WMMA_F32X4(a0h0, bB0, c);
        c = WMMA_F32X4(a0l4, bA4, c);
        c = WMMA_F32X4(a0h4, bB4, c);
        float h[8];
        ln_tanh(c, b0, g0, be0, h);    // all 32 lanes live

        // Layer 1: 8 -> 8
        {
            v2f pA0, pB0, pA4, pB4;
            hidden_b_frags<0>(h, lane, pA0, pB0);
            hidden_b_frags<4>(h, lane, pA4, pB4);
            v8f c1 = {};
            c1 = WMMA_F32X4(a1l0, pA0, c1);
            c1 = WMMA_F32X4(a1h0, pB0, c1);
            c1 = WMMA_F32X4(a1l4, pA4, c1);
            c1 = WMMA_F32X4(a1h4, pB4, c1);
            ln_tanh(c1, b1, g1, be1, h);
        }

        // Layer 2: 8 -> 8
        {
            v2f pA0, pB0, pA4, pB4;
            hidden_b_frags<0>(h, lane, pA0, pB0);
            hidden_b_frags<4>(h, lane, pA4, pB4);
            v8f c2 = {};
            c2 = WMMA_F32X4(a2l0, pA0, c2);
            c2 = WMMA_F32X4(a2h0, pB0, c2);
            c2 = WMMA_F32X4(a2l4, pA4, c2);
            c2 = WMMA_F32X4(a2h4, pB4, c2);
            ln_tanh(c2, b2, g2, be2, h);
        }

        // Output layer: 8 -> 1 per lane; contiguous 32-lane streaming store
        float acc = b3v[0];
#pragma unroll
        for (int f = 0; f < 8; ++f) acc += h[f] * W3[f];

        if (e < n_edges) __builtin_nontemporal_store(acc, out + e);
    }
}

extern "C" void kernel_launch(void* const* d_in, const int* in_sizes, int n_in,
                              void* d_out, int out_size, void* d_ws, size_t ws_size,
                              hipStream_t stream) {
    const float*     x   = (const float*)d_in[0];
    const long long* ei  = (const long long*)d_in[1];
    const float*     W0  = (const float*)d_in[2];
    const float*     b0  = (const float*)d_in[3];
    const float*     g0  = (const float*)d_in[4];
    const float*     be0 = (const float*)d_in[5];
    const float*     W1  = (const float*)d_in[6];
    const float*     b1  = (const float*)d_in[7];
    const float*     g1  = (const float*)d_in[8];
    const float*     be1 = (const float*)d_in[9];
    const float*     W2  = (const float*)d_in[10];
    const float*     b2  = (const float*)d_in[11];
    const float*     g2  = (const float*)d_in[12];
    const float*     be2 = (const float*)d_in[13];
    const float*     W3  = (const float*)d_in[14];
    const float*     b3  = (const float*)d_in[15];
    float* out = (float*)d_out;

    const int E = out_size;                 // one output per edge
    const int nIters = (E + 31) >> 5;       // 32 edges per wave-iteration
    const int wavesPerBlock = 256 / 32;
    int blocks = 2048;                      // 16384 waves, ~7.6 iters each at E=4M
    if (blocks * wavesPerBlock > nIters)
        blocks = (nIters + wavesPerBlock - 1) / wavesPerBlock;
    if (blocks < 1) blocks = 1;

    edge_mlp_wmma_kernel<<<blocks, 256, 0, stream>>>(
        x, ei, W0, b0, g0, be0, W1, b1, g1, be1, W2, b2, g2, be2, W3, b3,
        out, E);
}